// NodeBlock_12120397709384
// MI455X (gfx1250) — compile-verified
//
#include <hip/hip_runtime.h>
#include <hip/hip_bf16.h>

#define N_NODES 50000
#define N_EDGES 800000
#define HDIM    128

typedef __attribute__((ext_vector_type(2))) float v2f;
typedef __attribute__((ext_vector_type(4))) float v4f;
typedef __attribute__((ext_vector_type(8))) float v8f;

constexpr int HSTRIDE = 132;          // LDS row stride (floats); 132 % 64 == 4 -> conflict-free frag reads
constexpr int WAVES_PER_BLOCK = 4;    // 4 waves -> 33.8 KB static LDS (< 64 KB static limit)

// ---------------------------------------------------------------- zero scratch
__global__ __launch_bounds__(256) void zero_kernel(float* __restrict__ p, long long n) {
  long long i = (long long)blockIdx.x * blockDim.x + threadIdx.x;
  const long long stride = (long long)gridDim.x * blockDim.x;
  for (; i < n; i += stride) p[i] = 0.0f;
}

// ------------------------------------------------- scatter-add edges -> agg[N,H]
// One thread per (edge, 4-column group): float4 NT load + 4 f32 atomics (agg is L2-resident).
__global__ __launch_bounds__(256) void scatter_add_kernel(
    const float* __restrict__ edge_attr, const int* __restrict__ edge_index,
    float* __restrict__ agg) {
  const long long total = (long long)N_EDGES * (HDIM / 4);
  long long i = (long long)blockIdx.x * blockDim.x + threadIdx.x;
  const long long stride = (long long)gridDim.x * blockDim.x;
  for (; i < total; i += stride) {
    const long long e = i >> 5;                    // edge id
    const int g = ((int)(i & 31)) << 2;            // column group * 4
    const int recv = edge_index[N_EDGES + e];      // receivers = edge_index[1]
    v4f v = __builtin_nontemporal_load((const v4f*)(edge_attr + e * HDIM + g));
    float* dst = agg + (long long)recv * HDIM + g;
    atomicAdd(dst + 0, v.x);
    atomicAdd(dst + 1, v.y);
    atomicAdd(dst + 2, v.z);
    atomicAdd(dst + 3, v.w);
  }
}

// ------------------------------------------------- column sums of agg (for mean)
__global__ __launch_bounds__(256) void colsum_kernel(
    const float* __restrict__ agg, float* __restrict__ colsum) {
  __shared__ float s[256];
  const int t = threadIdx.x;
  const int col = t & (HDIM - 1);
  float sum = 0.0f;
  for (int r = blockIdx.x * 2 + (t >> 7); r < N_NODES; r += gridDim.x * 2)
    sum += agg[(long long)r * HDIM + col];
  s[t] = sum;
  __syncthreads();
  if (t < HDIM) atomicAdd(&colsum[col], s[t] + s[t + HDIM]);
}

// ------------------------------------------------- fused MLP via V_WMMA_F32_16X16X4_F32
// x = relu([node_attr | agg - mean] @ W1 + b1) @ W2 + b2
// One wave per 16-row tile. A frag (16x4 f32): lanes 0-15 carry K={k,k+1}, lanes 16-31 K={k+2,k+3}.
__global__ __launch_bounds__(32 * WAVES_PER_BLOCK) void mlp_wmma_kernel(
    const float* __restrict__ node_attr, const float* __restrict__ agg,
    const float* __restrict__ colsum,
    const float* __restrict__ W1, const float* __restrict__ b1,
    const float* __restrict__ W2, const float* __restrict__ b2,
    float* __restrict__ x_out) {
  __shared__ float hbuf[WAVES_PER_BLOCK][16 * HSTRIDE];
  const int wave = threadIdx.x >> 5;
  const int lane = threadIdx.x & 31;
  const int half = lane >> 4;     // 0: K pair {0,1}; 1: K pair {2,3}
  const int lr   = lane & 15;     // M row (A) / N col (B,C,D)
  const int tile = blockIdx.x * WAVES_PER_BLOCK + wave;
  const int nTiles = N_NODES / 16;           // 3125 (exact)
  const float inv_n = 1.0f / (float)N_NODES;
  float* hw = &hbuf[wave][0];

  if (tile < nTiles) {                       // wave-uniform: EXEC all-1 inside
    const int row = tile * 16 + lr;
    const float* na_row = node_attr + (long long)row * HDIM;
    const float* ag_row = agg       + (long long)row * HDIM;

    for (int nb = 0; nb < HDIM; nb += 16) {
      const int col = nb + lr;
      const float bias = b1[col];            // element (M,N) bias depends only on N
      v8f acc = {bias, bias, bias, bias, bias, bias, bias, bias};

      // K = 0..127 : node_attr half of the concat
#pragma unroll 4
      for (int k0 = 0; k0 < 128; k0 += 4) {
        const int kk = k0 + 2 * half;
        v2f a = *(const v2f*)(na_row + kk);
        v2f b;
        b.x = W1[(kk + 0) * HDIM + col];
        b.y = W1[(kk + 1) * HDIM + col];
        acc = __builtin_amdgcn_wmma_f32_16x16x4_f32(false, a, false, b,
                                                    (short)0, acc, false, false);
      }
      // K = 128..255 : (agg - mean) half of the concat
#pragma unroll 4
      for (int k0 = 0; k0 < 128; k0 += 4) {
        const int kk = k0 + 2 * half;
        v2f a = *(const v2f*)(ag_row + kk);
        v2f m = *(const v2f*)(colsum + kk);
        a.x -= m.x * inv_n;
        a.y -= m.y * inv_n;
        v2f b;
        b.x = W1[(128 + kk + 0) * HDIM + col];
        b.y = W1[(128 + kk + 1) * HDIM + col];
        acc = __builtin_amdgcn_wmma_f32_16x16x4_f32(false, a, false, b,
                                                    (short)0, acc, false, false);
      }
      // ReLU + transpose through LDS (D layout: VGPR v, lane -> M = v + 8*half, N = lr)
#pragma unroll
      for (int v = 0; v < 8; ++v) {
        float hv = acc[v];
        hv = hv > 0.0f ? hv : 0.0f;
        hw[(v + 8 * half) * HSTRIDE + col] = hv;
      }
    }
  }
  __syncthreads();
  if (tile < nTiles) {
    for (int nb = 0; nb < HDIM; nb += 16) {
      const int col = nb + lr;
      const float bias = b2[col];
      v8f acc = {bias, bias, bias, bias, bias, bias, bias, bias};
#pragma unroll 4
      for (int k0 = 0; k0 < 128; k0 += 4) {
        const int kk = k0 + 2 * half;
        v2f a = *(const v2f*)(hw + lr * HSTRIDE + kk);   // stride 132 -> conflict-free
        v2f b;
        b.x = W2[(kk + 0) * HDIM + col];
        b.y = W2[(kk + 1) * HDIM + col];
        acc = __builtin_amdgcn_wmma_f32_16x16x4_f32(false, a, false, b,
                                                    (short)0, acc, false, false);
      }
#pragma unroll
      for (int v = 0; v < 8; ++v)
        x_out[(long long)(tile * 16 + v + 8 * half) * HDIM + col] = acc[v];
    }
  }
}

// ------------------------------------------------- tuple tail: edge_index, edge_attr
__global__ __launch_bounds__(256) void copy_ei_kernel(
    const int* __restrict__ ei, float* __restrict__ out, long long base, long long limit) {
  const long long total = 2LL * N_EDGES;
  long long i = (long long)blockIdx.x * blockDim.x + threadIdx.x;
  const long long stride = (long long)gridDim.x * blockDim.x;
  for (; i < total; i += stride) {
    const long long o = base + i;
    if (o < limit) out[o] = (float)ei[i];
  }
}

__global__ __launch_bounds__(256) void copy_ea_kernel(
    const float* __restrict__ ea, float* __restrict__ out, long long base, long long limit) {
  const long long total = (long long)N_EDGES * (HDIM / 4);
  long long i = (long long)blockIdx.x * blockDim.x + threadIdx.x;
  const long long stride = (long long)gridDim.x * blockDim.x;
  for (; i < total; i += stride) {
    const long long o = base + i * 4;
    if (o + 3 < limit) {
      v4f v = __builtin_nontemporal_load((const v4f*)(ea + i * 4));
      __builtin_nontemporal_store(v, (v4f*)(out + o));
    }
  }
}

// ---------------------------------------------------------------- launcher
extern "C" void kernel_launch(void* const* d_in, const int* in_sizes, int n_in,
                              void* d_out, int out_size, void* d_ws, size_t ws_size,
                              hipStream_t stream) {
  const float* node_attr  = (const float*)d_in[0];
  const int*   edge_index = (const int*)d_in[1];
  const float* edge_attr  = (const float*)d_in[2];
  const float* W1 = (const float*)d_in[3];
  const float* b1 = (const float*)d_in[4];
  const float* W2 = (const float*)d_in[5];
  const float* b2 = (const float*)d_in[6];
  float* out = (float*)d_out;

  float* agg    = (float*)d_ws;                       // N_NODES*HDIM floats
  float* colsum = agg + (long long)N_NODES * HDIM;    // HDIM floats

  const long long zero_n = (long long)N_NODES * HDIM + HDIM;
  zero_kernel<<<2048, 256, 0, stream>>>(agg, zero_n);
  scatter_add_kernel<<<8192, 256, 0, stream>>>(edge_attr, edge_index, agg);
  colsum_kernel<<<512, 256, 0, stream>>>(agg, colsum);

  const int nTiles = N_NODES / 16;
  const int blocks = (nTiles + WAVES_PER_BLOCK - 1) / WAVES_PER_BLOCK;
  mlp_wmma_kernel<<<blocks, 32 * WAVES_PER_BLOCK, 0, stream>>>(
      node_attr, agg, colsum, W1, b1, W2, b2, out);

  const long long X  = (long long)N_NODES * HDIM;     // x output elements
  const long long EI = 2LL * N_EDGES;
  copy_ei_kernel<<<1024, 256, 0, stream>>>(edge_index, out, X, (long long)out_size);
  copy_ea_kernel<<<8192, 256, 0, stream>>>(edge_attr, out, X + EI, (long long)out_size);
}